// GCN_40836549050952
// MI455X (gfx1250) — compile-verified
//
#include <hip/hip_runtime.h>

typedef __attribute__((ext_vector_type(2))) float v2f;
typedef __attribute__((ext_vector_type(8))) float v8f;

#define HDIM 64
#define BN_EPS 1e-5f

// ---------------- degree / dinv ----------------
__global__ void deg_init_kernel(float* __restrict__ deg, int n) {
    int i = blockIdx.x * blockDim.x + threadIdx.x;
    if (i < n) deg[i] = 1.0f;  // self loop
}

__global__ void deg_count_kernel(const int* __restrict__ dst, float* __restrict__ deg, int e) {
    int i = blockIdx.x * blockDim.x + threadIdx.x;
    if (i < e) atomicAdd(&deg[dst[i]], 1.0f);
}

__global__ void deg_rsqrt_kernel(float* __restrict__ deg, int n) {
    int i = blockIdx.x * blockDim.x + threadIdx.x;
    if (i < n) deg[i] = rsqrtf(deg[i]);
}

// ---------------- WMMA f32 GEMM: y = (x @ W^T + b) * dinv[row]; acc = y ----------------
// x: [nrows, K] row-major, W: [64, K] row-major, per-wave 16x64 output strip.
template <int K>
__global__ __launch_bounds__(256) void gemm_wmma_kernel(
    const float* __restrict__ x, const float* __restrict__ W,
    const float* __restrict__ bias, const float* __restrict__ dinv,
    float* __restrict__ y, float* __restrict__ acc, int nrows) {
    __shared__ float sW[HDIM * K];
    for (int i = threadIdx.x; i < HDIM * K; i += blockDim.x) sW[i] = W[i];
    __syncthreads();

    const int wave = threadIdx.x >> 5;
    const int lane = threadIdx.x & 31;
    const int tile = blockIdx.x * 8 + wave;
    const int ntiles = (nrows + 15) >> 4;
    if (tile >= ntiles) return;  // wave-uniform: EXEC all-1s inside

    const int m_lo = lane & 15;
    const int n_lo = lane & 15;
    const int k_half = (lane >> 4) << 1;  // 0 or 2 (A/B VGPR layout for 16x16x4 f32)

    int row = tile * 16 + m_lo;
    if (row >= nrows) row = nrows - 1;  // clamped read (stores are guarded)
    const float* xr = x + (size_t)row * K;

    v8f c0 = {}, c1 = {}, c2 = {}, c3 = {};
    for (int k0 = 0; k0 < K; k0 += 4) {
        const int kk = k0 + k_half;
        v2f a;
        a.x = xr[kk];
        a.y = xr[kk + 1];
        v2f b0, b1, b2, b3;  // B[k][n] = W[n][k]
        b0.x = sW[(n_lo +  0) * K + kk]; b0.y = sW[(n_lo +  0) * K + kk + 1];
        b1.x = sW[(n_lo + 16) * K + kk]; b1.y = sW[(n_lo + 16) * K + kk + 1];
        b2.x = sW[(n_lo + 32) * K + kk]; b2.y = sW[(n_lo + 32) * K + kk + 1];
        b3.x = sW[(n_lo + 48) * K + kk]; b3.y = sW[(n_lo + 48) * K + kk + 1];
        c0 = __builtin_amdgcn_wmma_f32_16x16x4_f32(false, a, false, b0, (short)0, c0, false, false);
        c1 = __builtin_amdgcn_wmma_f32_16x16x4_f32(false, a, false, b1, (short)0, c1, false, false);
        c2 = __builtin_amdgcn_wmma_f32_16x16x4_f32(false, a, false, b2, (short)0, c2, false, false);
        c3 = __builtin_amdgcn_wmma_f32_16x16x4_f32(false, a, false, b3, (short)0, c3, false, false);
    }

    const float bb0 = bias[n_lo], bb1 = bias[16 + n_lo];
    const float bb2 = bias[32 + n_lo], bb3 = bias[48 + n_lo];
    const int r_off = (lane >> 4) << 3;  // C/D layout: upper half-lanes hold rows +8
    for (int g = 0; g < 8; ++g) {
        const int m = tile * 16 + r_off + g;
        if (m < nrows) {
            const float dv = dinv[m];
            const size_t base = (size_t)m * HDIM;
            const float v0 = (c0[g] + bb0) * dv;
            const float v1 = (c1[g] + bb1) * dv;
            const float v2 = (c2[g] + bb2) * dv;
            const float v3 = (c3[g] + bb3) * dv;
            y[base + n_lo] = v0;  y[base + 16 + n_lo] = v1;
            y[base + 32 + n_lo] = v2;  y[base + 48 + n_lo] = v3;
            acc[base + n_lo] = v0;  acc[base + 16 + n_lo] = v1;
            acc[base + 32 + n_lo] = v2;  acc[base + 48 + n_lo] = v3;
        }
    }
}

// ---------------- edge scatter: acc[dst] += y[src] (one wave per edge) ----------------
__global__ void scatter_kernel(const int* __restrict__ src, const int* __restrict__ dst,
                               const float* __restrict__ y, float* __restrict__ acc, int e) {
    const long long t = (long long)blockIdx.x * blockDim.x + threadIdx.x;
    const int edge = (int)(t >> 5);
    const int lane = (int)(t & 31);
    if (edge < e) {
        const int s = src[edge];
        const int d = dst[edge];
        const float2 v = *(const float2*)(y + (size_t)s * HDIM + lane * 2);
        float* ap = acc + (size_t)d * HDIM + lane * 2;
        atomicAdd(ap, v.x);
        atomicAdd(ap + 1, v.y);
    }
}

// ---------------- BN statistics ----------------
__global__ void zero_stats_kernel(float* __restrict__ sums) {
    if (threadIdx.x < 128) sums[threadIdx.x] = 0.0f;
}

__global__ void stats_kernel(const float* __restrict__ acc, const float* __restrict__ dinv,
                             float* __restrict__ sums, long long total) {
    __shared__ float ls[128];
    for (int i = threadIdx.x; i < 128; i += blockDim.x) ls[i] = 0.0f;
    __syncthreads();
    const long long stride = (long long)gridDim.x * blockDim.x;
    for (long long i = (long long)blockIdx.x * blockDim.x + threadIdx.x; i < total; i += stride) {
        const int col = (int)(i & 63);
        const long long row = i >> 6;
        const float v = acc[i] * dinv[row];
        atomicAdd(&ls[col], v);
        atomicAdd(&ls[64 + col], v * v);
    }
    __syncthreads();
    for (int i = threadIdx.x; i < 128; i += blockDim.x) atomicAdd(&sums[i], ls[i]);
}

// ---------------- BN apply + ReLU: h = relu(((acc*dinv)-mu)*rsqrt(var+eps)*g + be) ----------------
__global__ void bn_apply_kernel(const float* __restrict__ acc, const float* __restrict__ dinv,
                                const float* __restrict__ sums, const float* __restrict__ gamma,
                                const float* __restrict__ beta, float* __restrict__ outh,
                                long long total, float invN) {
    const long long i = (long long)blockIdx.x * blockDim.x + threadIdx.x;
    if (i < total) {
        const int col = (int)(i & 63);
        const long long row = i >> 6;
        const float mean = sums[col] * invN;
        const float var = sums[64 + col] * invN - mean * mean;
        const float h = acc[i] * dinv[row];
        const float v = (h - mean) * rsqrtf(var + BN_EPS) * gamma[col] + beta[col];
        outh[i] = v > 0.0f ? v : 0.0f;
    }
}

// ---------------- log_softmax over 64 cols, one wave per row, in-place safe ----------------
__global__ void log_softmax_kernel(const float* __restrict__ acc, const float* __restrict__ dinv,
                                   float* __restrict__ out, int n) {
    const long long t = (long long)blockIdx.x * blockDim.x + threadIdx.x;
    const int row = (int)(t >> 5);
    const int lane = (int)(t & 31);
    if (row < n) {
        const float dv = dinv[row];
        const size_t base = (size_t)row * HDIM;
        const float z0 = acc[base + lane] * dv;
        const float z1 = acc[base + 32 + lane] * dv;
        float m = fmaxf(z0, z1);
        for (int o = 16; o > 0; o >>= 1) m = fmaxf(m, __shfl_xor(m, o, 32));
        float s = expf(z0 - m) + expf(z1 - m);
        for (int o = 16; o > 0; o >>= 1) s += __shfl_xor(s, o, 32);
        const float l = m + logf(s);
        out[base + lane] = z0 - l;
        out[base + 32 + lane] = z1 - l;
    }
}

extern "C" void kernel_launch(void* const* d_in, const int* in_sizes, int n_in,
                              void* d_out, int out_size, void* d_ws, size_t ws_size,
                              hipStream_t stream) {
    const float* x   = (const float*)d_in[0];
    const int*   ei  = (const int*)d_in[1];
    const float* W1  = (const float*)d_in[2];
    const float* b1  = (const float*)d_in[3];
    const float* g1  = (const float*)d_in[4];
    const float* be1 = (const float*)d_in[5];
    const float* W2  = (const float*)d_in[6];
    const float* b2  = (const float*)d_in[7];
    const float* g2  = (const float*)d_in[8];
    const float* be2 = (const float*)d_in[9];
    const float* W3  = (const float*)d_in[10];
    const float* b3  = (const float*)d_in[11];

    const int N = in_sizes[0] / 128;
    const int E = in_sizes[1] / 2;
    const int* src = ei;
    const int* dst = ei + E;
    float* out = (float*)d_out;

    // workspace layout: dinv | bufA (y) | bufB (h) | sums[128]; acc aliases d_out
    float* ws = (float*)d_ws;
    const size_t NH = (size_t)N * HDIM;
    float* dinv = ws;
    float* bufA = ws + (((size_t)N + 255) & ~(size_t)255);
    float* bufB = bufA + NH;
    float* sums = bufB + NH;
    float* acc = out;

    const long long total = (long long)N * HDIM;
    const int nblk = (N + 255) / 256;
    const int eblk = (E + 255) / 256;
    const int ntiles = (N + 15) / 16;
    const int gblk = (ntiles + 7) / 8;
    const int sblk = (int)(((long long)E * 32 + 255) / 256);
    const int ablk = (int)((total + 255) / 256);
    const int lblk = (int)(((long long)N * 32 + 255) / 256);
    const float invN = 1.0f / (float)N;

    // degrees -> dinv
    deg_init_kernel<<<nblk, 256, 0, stream>>>(dinv, N);
    deg_count_kernel<<<eblk, 256, 0, stream>>>(dst, dinv, E);
    deg_rsqrt_kernel<<<nblk, 256, 0, stream>>>(dinv, N);

    // ---- layer 1 (K=128) ----
    gemm_wmma_kernel<128><<<gblk, 256, 0, stream>>>(x, W1, b1, dinv, bufA, acc, N);
    scatter_kernel<<<sblk, 256, 0, stream>>>(src, dst, bufA, acc, E);
    zero_stats_kernel<<<1, 128, 0, stream>>>(sums);
    stats_kernel<<<1024, 256, 0, stream>>>(acc, dinv, sums, total);
    bn_apply_kernel<<<ablk, 256, 0, stream>>>(acc, dinv, sums, g1, be1, bufB, total, invN);

    // ---- layer 2 (K=64) ----
    gemm_wmma_kernel<64><<<gblk, 256, 0, stream>>>(bufB, W2, b2, dinv, bufA, acc, N);
    scatter_kernel<<<sblk, 256, 0, stream>>>(src, dst, bufA, acc, E);
    zero_stats_kernel<<<1, 128, 0, stream>>>(sums);
    stats_kernel<<<1024, 256, 0, stream>>>(acc, dinv, sums, total);
    bn_apply_kernel<<<ablk, 256, 0, stream>>>(acc, dinv, sums, g2, be2, bufB, total, invN);

    // ---- layer 3 (K=64) + log_softmax ----
    gemm_wmma_kernel<64><<<gblk, 256, 0, stream>>>(bufB, W3, b3, dinv, bufA, acc, N);
    scatter_kernel<<<sblk, 256, 0, stream>>>(src, dst, bufA, acc, E);
    log_softmax_kernel<<<lblk, 256, 0, stream>>>(acc, dinv, out, N);
}